// HEATlayer_61194694034245
// MI455X (gfx1250) — compile-verified
//
#include <hip/hip_runtime.h>
#include <hip/hip_bf16.h>

typedef __attribute__((ext_vector_type(2))) float v2f;
typedef __attribute__((ext_vector_type(8))) float v8f;

#define NN 1536
#define EE 49152
#define SLOPE 0.2f

__device__ __forceinline__ float leaky(float x) { return x >= 0.f ? x : SLOPE * x; }

// order-preserving float<->uint encoding for atomicMax over signed floats
__device__ __forceinline__ unsigned encf(float f) {
    unsigned u = __float_as_uint(f);
    return (u & 0x80000000u) ? ~u : (u | 0x80000000u);
}
__device__ __forceinline__ float decf(unsigned u) {
    return (u & 0x80000000u) ? __uint_as_float(u & 0x7FFFFFFFu) : __uint_as_float(~u);
}

__device__ __forceinline__ v8f wmma4(v2f a, v2f b, v8f c) {
    // D = A(16x4 f32) * B(4x16 f32) + C(16x16 f32)
    return __builtin_amdgcn_wmma_f32_16x16x4_f32(false, a, false, b, (short)0, c, false, false);
}

// ---------------------------------------------------------------------------
// K1: per 16-node tile (one wave32):
//   emb = leaky( node_f @ (veh ? W_veh : W_ped)^T )          [WMMA, K=32]
//   nbr_up = emb @ Wu_n^T                                    [WMMA, K=32]
//   tar = emb @ W_tar^T, nbr = emb @ W_nbr^T (3 heads, VALU)
//   colsum[o] += sum_rows leaky(nbr_up)   (fallback for edgeless rows)
// ---------------------------------------------------------------------------
__global__ void k_node(const float* __restrict__ node_f,
                       const unsigned char* __restrict__ veh_mask,
                       const float* __restrict__ W_veh,
                       const float* __restrict__ W_ped,
                       const float* __restrict__ W_att,
                       const float* __restrict__ W_upd,
                       float* __restrict__ nbr_up,
                       float* __restrict__ tarArr,
                       float* __restrict__ nbrArr,
                       float* __restrict__ colsum)
{
    __shared__ float sh[16][32];
    const int lane = threadIdx.x;
    const int m    = lane & 15;
    const int hi   = lane >> 4;           // 0: K/M-low half, 1: high half
    const int base = blockIdx.x * 16;

    v8f cV0 = {}, cV1 = {}, cP0 = {}, cP1 = {};
    const float* xf = node_f + (size_t)(base + m) * 32;
#pragma unroll
    for (int kb = 0; kb < 32; kb += 4) {
        const int k0 = kb + hi * 2;
        v2f a;  a.x = xf[k0];                    a.y = xf[k0 + 1];
        v2f bv0; bv0.x = W_veh[m * 32 + k0];     bv0.y = W_veh[m * 32 + k0 + 1];
        v2f bv1; bv1.x = W_veh[(m + 16) * 32 + k0]; bv1.y = W_veh[(m + 16) * 32 + k0 + 1];
        v2f bp0; bp0.x = W_ped[m * 32 + k0];     bp0.y = W_ped[m * 32 + k0 + 1];
        v2f bp1; bp1.x = W_ped[(m + 16) * 32 + k0]; bp1.y = W_ped[(m + 16) * 32 + k0 + 1];
        cV0 = wmma4(a, bv0, cV0);
        cV1 = wmma4(a, bv1, cV1);
        cP0 = wmma4(a, bp0, cP0);
        cP1 = wmma4(a, bp1, cP1);
    }
    // select per node (C layout: lane holds col N=m, VGPR r -> row M = r + hi*8)
#pragma unroll
    for (int r = 0; r < 8; ++r) {
        const int row = r + hi * 8;
        const bool veh = veh_mask[base + row] != 0;
        sh[row][m]      = leaky(veh ? cV0[r] : cP0[r]);
        sh[row][m + 16] = leaky(veh ? cV1[r] : cP1[r]);
    }
    __syncthreads();

    // nbr_up = emb @ Wu_n^T   (Wu_n = W_upd[:, 32:64])
    v8f u0 = {}, u1 = {};
#pragma unroll
    for (int kb = 0; kb < 32; kb += 4) {
        const int k0 = kb + hi * 2;
        v2f a;  a.x = sh[m][k0];  a.y = sh[m][k0 + 1];
        v2f b0; b0.x = W_upd[m * 64 + 32 + k0];        b0.y = W_upd[m * 64 + 32 + k0 + 1];
        v2f b1; b1.x = W_upd[(m + 16) * 64 + 32 + k0]; b1.y = W_upd[(m + 16) * 64 + 32 + k0 + 1];
        u0 = wmma4(a, b0, u0);
        u1 = wmma4(a, b1, u1);
    }
    float s0 = 0.f, s1 = 0.f;
#pragma unroll
    for (int r = 0; r < 8; ++r) {
        const int row = base + r + hi * 8;
        nbr_up[row * 32 + m]      = u0[r];
        nbr_up[row * 32 + 16 + m] = u1[r];
        s0 += leaky(u0[r]);
        s1 += leaky(u1[r]);
    }
    atomicAdd(&colsum[m], s0);
    atomicAdd(&colsum[16 + m], s1);

    // tar (lanes 0-15) / nbr (lanes 16-31): 3 heads, 32-dot each
    if (hi == 0) {
#pragma unroll
        for (int h = 0; h < 3; ++h) {
            float s = 0.f;
            for (int c = 0; c < 32; ++c) s += sh[m][c] * W_att[h * 128 + c];
            tarArr[(base + m) * 3 + h] = s;
        }
    } else {
#pragma unroll
        for (int h = 0; h < 3; ++h) {
            float s = 0.f;
            for (int c = 0; c < 32; ++c) s += sh[m][c] * W_att[h * 128 + 96 + c];
            nbrArr[(base + m) * 3 + h] = s;
        }
    }
}

// ---------------------------------------------------------------------------
// K2: per-edge embeddings  embA = leaky(edge_attr @ W_ea^T), embT likewise
// thread = (e, o)
// ---------------------------------------------------------------------------
__global__ void k_edge_emb(const float* __restrict__ edge_attr,
                           const float* __restrict__ edge_type,
                           const float* __restrict__ W_ea,
                           const float* __restrict__ W_et,
                           float* __restrict__ embA,
                           float* __restrict__ embT)
{
    const int idx = blockIdx.x * blockDim.x + threadIdx.x;
    if (idx >= EE * 32) return;
    const int e = idx >> 5;
    const int o = idx & 31;
    const float a0 = edge_attr[e * 2], a1 = edge_attr[e * 2 + 1];
    const float t0 = edge_type[e * 2], t1 = edge_type[e * 2 + 1];
    embA[idx] = leaky(a0 * W_ea[o * 2] + a1 * W_ea[o * 2 + 1]);
    embT[idx] = leaky(t0 * W_et[o * 2] + t1 * W_et[o * 2 + 1]);
}

// ---------------------------------------------------------------------------
// K3: deduplicate (src,dst) pairs: first edge claims slot, duplicates merge
// their features into the representative (sum happens BEFORE the leaky in
// scores/upd, matching the dense scatter-add semantics).
// ---------------------------------------------------------------------------
__global__ void k_dedup(const int* __restrict__ edge_index,
                        int* __restrict__ owner,
                        int* __restrict__ isRep,
                        float* __restrict__ embA,
                        float* __restrict__ embT)
{
    const int e = blockIdx.x * blockDim.x + threadIdx.x;
    if (e >= EE) return;
    const int src = edge_index[e];
    const int dst = edge_index[EE + e];
    const int old = atomicCAS(&owner[src * NN + dst], -1, e);
    if (old < 0) {
        isRep[e] = 1;
    } else {
        isRep[e] = 0;
        for (int o = 0; o < 32; ++o) {
            atomicAdd(&embA[old * 32 + o], embA[e * 32 + o]);
            atomicAdd(&embT[old * 32 + o], embT[e * 32 + o]);
        }
    }
}

// ---------------------------------------------------------------------------
// K4: U = embA @ Wu_e^T   (E x 32 @ 32 x 32) — WMMA, one wave per 16 edges.
// Runs after dedup so representatives see merged features.
// ---------------------------------------------------------------------------
__global__ void k_upd_gemm(const float* __restrict__ embA,
                           const float* __restrict__ W_upd,
                           float* __restrict__ U)
{
    const int lane = threadIdx.x;
    const int m    = lane & 15;
    const int hi   = lane >> 4;
    const int base = blockIdx.x * 16;

    v8f c0 = {}, c1 = {};
    const float* af = embA + (size_t)(base + m) * 32;
#pragma unroll
    for (int kb = 0; kb < 32; kb += 4) {
        const int k0 = kb + hi * 2;
        v2f a;  a.x = af[k0];  a.y = af[k0 + 1];
        v2f b0; b0.x = W_upd[m * 64 + k0];        b0.y = W_upd[m * 64 + k0 + 1];
        v2f b1; b1.x = W_upd[(m + 16) * 64 + k0]; b1.y = W_upd[(m + 16) * 64 + k0 + 1];
        c0 = wmma4(a, b0, c0);
        c1 = wmma4(a, b1, c1);
    }
#pragma unroll
    for (int r = 0; r < 8; ++r) {
        const int row = base + r + hi * 8;
        U[row * 32 + m]      = c0[r];
        U[row * 32 + 16 + m] = c1[r];
    }
}

// ---------------------------------------------------------------------------
// K5: per representative edge: leaky attention score (3 heads) + row max
// ---------------------------------------------------------------------------
__global__ void k_score(const int* __restrict__ edge_index,
                        const int* __restrict__ isRep,
                        const float* __restrict__ embA,
                        const float* __restrict__ embT,
                        const float* __restrict__ W_att,
                        const float* __restrict__ tarArr,
                        const float* __restrict__ nbrArr,
                        float* __restrict__ sc,
                        unsigned* __restrict__ rowmaxU)
{
    const int e = blockIdx.x * blockDim.x + threadIdx.x;
    if (e >= EE) return;
    if (!isRep[e]) return;
    const int src = edge_index[e];
    const int dst = edge_index[EE + e];
    float acc[3];
#pragma unroll
    for (int h = 0; h < 3; ++h) acc[h] = tarArr[src * 3 + h] + nbrArr[dst * 3 + h];
    for (int c = 0; c < 32; ++c) {
        const float ea = embA[e * 32 + c];
        const float et = embT[e * 32 + c];
#pragma unroll
        for (int h = 0; h < 3; ++h)
            acc[h] += ea * W_att[h * 128 + 32 + c] + et * W_att[h * 128 + 64 + c];
    }
#pragma unroll
    for (int h = 0; h < 3; ++h) {
        const float s = leaky(acc[h]);
        sc[e * 3 + h] = s;
        atomicMax(&rowmaxU[src * 3 + h], encf(s));
    }
}

// ---------------------------------------------------------------------------
// K6: softmax numerator accumulation: 32 lanes per edge (one per output o)
//   upd = leaky(U[e] + nbr_up[dst]);  num[src,h,o] += w_h * upd_o
// ---------------------------------------------------------------------------
__global__ void k_agg(const int* __restrict__ edge_index,
                      const int* __restrict__ isRep,
                      const float* __restrict__ sc,
                      const unsigned* __restrict__ rowmaxU,
                      const float* __restrict__ U,
                      const float* __restrict__ nbr_up,
                      float* __restrict__ num,
                      float* __restrict__ rowsum)
{
    const int t = blockIdx.x * blockDim.x + threadIdx.x;
    const int e = t >> 5;
    const int o = t & 31;
    if (e >= EE) return;
    if (!isRep[e]) return;
    const int src = edge_index[e];
    const int dst = edge_index[EE + e];
    const float w0 = __expf(sc[e * 3 + 0] - decf(rowmaxU[src * 3 + 0]));
    const float w1 = __expf(sc[e * 3 + 1] - decf(rowmaxU[src * 3 + 1]));
    const float w2 = __expf(sc[e * 3 + 2] - decf(rowmaxU[src * 3 + 2]));
    const float updv = leaky(U[e * 32 + o] + nbr_up[dst * 32 + o]);
    atomicAdd(&num[src * 96 + o],      w0 * updv);
    atomicAdd(&num[src * 96 + 32 + o], w1 * updv);
    atomicAdd(&num[src * 96 + 64 + o], w2 * updv);
    if (o == 0) {
        atomicAdd(&rowsum[src * 3 + 0], w0);
        atomicAdd(&rowsum[src * 3 + 1], w1);
        atomicAdd(&rowsum[src * 3 + 2], w2);
    }
}

// ---------------------------------------------------------------------------
// K7: out[i, h*32+o] = num/rowsum, or colmean fallback for edgeless rows
// ---------------------------------------------------------------------------
__global__ void k_out(const float* __restrict__ num,
                      const float* __restrict__ rowsum,
                      const float* __restrict__ colsum,
                      float* __restrict__ out)
{
    const int t = blockIdx.x * blockDim.x + threadIdx.x;
    if (t >= NN * 96) return;
    const int i  = t / 96;
    const int ho = t % 96;
    const int h  = ho >> 5;
    const int o  = ho & 31;
    const float rs = rowsum[i * 3 + h];
    out[t] = (rs > 0.f) ? (num[t] / rs) : (colsum[o] * (1.0f / NN));
}

// ---------------------------------------------------------------------------
extern "C" void kernel_launch(void* const* d_in, const int* in_sizes, int n_in,
                              void* d_out, int out_size, void* d_ws, size_t ws_size,
                              hipStream_t stream)
{
    const float* node_f    = (const float*)d_in[0];
    const float* edge_attr = (const float*)d_in[1];
    const float* edge_type = (const float*)d_in[2];
    const float* W_veh     = (const float*)d_in[3];
    const float* W_ped     = (const float*)d_in[4];
    const float* W_ea      = (const float*)d_in[5];
    const float* W_et      = (const float*)d_in[6];
    const float* W_att     = (const float*)d_in[7];
    const float* W_upd     = (const float*)d_in[8];
    const int*   edge_index = (const int*)d_in[9];
    const unsigned char* veh_mask = (const unsigned char*)d_in[10];
    float* out = (float*)d_out;

    char* ws = (char*)d_ws;
    size_t off = 0;
    auto take = [&](size_t bytes) -> char* {
        char* p = ws + off;
        off += (bytes + 255) & ~(size_t)255;
        return p;
    };
    float*    nbr_up  = (float*)take((size_t)NN * 32 * 4);
    float*    tarArr  = (float*)take((size_t)NN * 3 * 4);
    float*    nbrArr  = (float*)take((size_t)NN * 3 * 4);
    float*    embA    = (float*)take((size_t)EE * 32 * 4);
    float*    embT    = (float*)take((size_t)EE * 32 * 4);
    float*    U       = (float*)take((size_t)EE * 32 * 4);
    float*    sc      = (float*)take((size_t)EE * 3 * 4);
    int*      isRep   = (int*)take((size_t)EE * 4);
    int*      owner   = (int*)take((size_t)NN * NN * 4);
    // zero-initialized region (contiguous): rowmaxU | rowsum | num | colsum
    size_t zoff = off;
    unsigned* rowmaxU = (unsigned*)take((size_t)NN * 3 * 4);
    float*    rowsum  = (float*)take((size_t)NN * 3 * 4);
    float*    num     = (float*)take((size_t)NN * 96 * 4);
    float*    colsum  = (float*)take(32 * 4);
    size_t zbytes = off - zoff;

    hipMemsetAsync(owner, 0xFF, (size_t)NN * NN * 4, stream);   // owner = -1
    hipMemsetAsync(ws + zoff, 0, zbytes, stream);               // maxes(-inf enc)/sums = 0

    k_node<<<NN / 16, 32, 0, stream>>>(node_f, veh_mask, W_veh, W_ped, W_att, W_upd,
                                       nbr_up, tarArr, nbrArr, colsum);
    k_edge_emb<<<(EE * 32) / 256, 256, 0, stream>>>(edge_attr, edge_type, W_ea, W_et,
                                                    embA, embT);
    k_dedup<<<EE / 256, 256, 0, stream>>>(edge_index, owner, isRep, embA, embT);
    k_upd_gemm<<<EE / 16, 32, 0, stream>>>(embA, W_upd, U);
    k_score<<<EE / 256, 256, 0, stream>>>(edge_index, isRep, embA, embT, W_att,
                                          tarArr, nbrArr, sc, rowmaxU);
    k_agg<<<(EE * 32) / 256, 256, 0, stream>>>(edge_index, isRep, sc, rowmaxU, U,
                                               nbr_up, num, rowsum);
    k_out<<<(NN * 96 + 255) / 256, 256, 0, stream>>>(num, rowsum, colsum, out);
}